// CRFLayer_21277267984575
// MI455X (gfx1250) — compile-verified
//
#include <hip/hip_runtime.h>
#include <hip/hip_bf16.h>
#include <stdint.h>

// ---------------------------------------------------------------------------
// Viterbi decode (CRF) for MI455X / gfx1250.
//   feats:       [8192][512] f32   transitions: [512][512] f32
//   out:         [0]=score(f32), [1..8192]=path (ints stored as f32)
//
// Max-plus semiring + argmax => WMMA not applicable (wrong algebra, no argmax).
// CDNA5 paths used: global_load_async_to_lds_b128/b32 + s_wait_asynccnt for
// LDS staging, scoped agent fences, global_prefetch, 128KB LDS-resident
// transition slices (320KB/WGP), wave32 sizing, device-scope L2 atomics for
// the 8-WGP cooperative scan, segmented pointer-jumping parallel backtrack.
//
// d_ws layout: [0,4)            int sync counter
//              [256,4352)       float scoresBuf[2][512] (double buffer)
//              [8192, 8395776)  u16 backPointers[8191][512]  (~8.39 MB)
//              [8395776, +64KB) u16 segMap[64][512]
// ---------------------------------------------------------------------------

#define NWG    8
#define NTHR   512
#define T_LEN  8192
#define L      512
#define COLS   64            // destination columns per workgroup (L/NWG)
#define CHUNKS 8             // threads cooperating per column (NTHR/COLS)
#define CLEN   64            // source labels per chunk (L/CHUNKS)
#define SEG    128           // backtrack segment length (rows)
#define NSEG   64            // ceil((T_LEN-1)/SEG) segments (last has 127 rows)
#define SEGS_PER_WG (NSEG / NWG)

static_assert(NTHR == L, "staging assumes one thread per label");
static_assert(COLS * CHUNKS == NTHR, "");
static_assert(CHUNKS * CLEN == L, "");
static_assert(NSEG * SEG == T_LEN, "");

#define AS1 __attribute__((address_space(1)))
#define AS3 __attribute__((address_space(3)))

typedef int vint4 __attribute__((vector_size(16)));   // matches builtin param type

#if defined(__has_builtin)
#if __has_builtin(__builtin_amdgcn_global_load_async_to_lds_b128) && \
    __has_builtin(__builtin_amdgcn_global_load_async_to_lds_b32)
#define HAVE_ASYNC_LDS 1
#endif
#endif
#ifndef HAVE_ASYNC_LDS
#define HAVE_ASYNC_LDS 0
#endif

// Generic->LDS pointer: ISA 10.2 aperture rule, LDS offset = addr[31:0].
__device__ __forceinline__ void async_copy_b128(const void* gp, void* lp) {
#if HAVE_ASYNC_LDS
  __builtin_amdgcn_global_load_async_to_lds_b128(
      (AS1 vint4*)(uintptr_t)gp,
      (AS3 vint4*)(uint32_t)(uintptr_t)lp, 0, 0);
#else
  *(float4*)lp = *(const float4*)gp;
#endif
}

__device__ __forceinline__ void async_copy_b32(const void* gp, void* lp) {
#if HAVE_ASYNC_LDS
  __builtin_amdgcn_global_load_async_to_lds_b32(
      (AS1 int*)(uintptr_t)gp,
      (AS3 int*)(uint32_t)(uintptr_t)lp, 0, 0);
#else
  *(float*)lp = *(const float*)gp;
#endif
}

__device__ __forceinline__ void wait_async_zero() {
#if HAVE_ASYNC_LDS
#if __has_builtin(__builtin_amdgcn_s_wait_asynccnt)
  __builtin_amdgcn_s_wait_asynccnt(0);
#else
  asm volatile("s_wait_asynccnt 0" ::: "memory");
#endif
#endif
}

__device__ __forceinline__ void fence_acquire_agent() {
#if __has_builtin(__builtin_amdgcn_fence)
  __builtin_amdgcn_fence(__ATOMIC_ACQUIRE, "agent");
#else
  __threadfence();
#endif
}
__device__ __forceinline__ void fence_release_agent() {
#if __has_builtin(__builtin_amdgcn_fence)
  __builtin_amdgcn_fence(__ATOMIC_RELEASE, "agent");
#else
  __threadfence();
#endif
}

__global__ void crf_init_kernel(int* cnt) {
  if (threadIdx.x == 0) *cnt = 0;
}

__global__ void __launch_bounds__(NTHR) crf_forward_kernel(
    const float* __restrict__ feats,
    const float* __restrict__ trans,
    float* __restrict__ out,
    int* __restrict__ cnt,
    float* __restrict__ scoresBuf,        // [2][L]
    unsigned short* __restrict__ bp,      // [T_LEN-1][L]
    unsigned short* __restrict__ segMap)  // [NSEG][L]
{
  extern __shared__ char smem[];
  float* ldsT      = (float*)smem;                               // [L][COLS] 128KB
  float* ldsScores = (float*)(smem + (size_t)L * COLS * 4);      // [L]
  float* redV      = (float*)(smem + (size_t)L * COLS * 4 + L * 4);
  int*   redI      = (int*)  (smem + (size_t)L * COLS * 4 + L * 4 + CHUNKS * COLS * 4);

  const int tid = threadIdx.x;
  const int wg  = blockIdx.x;

  // ---- one-time: transitions slice [i=0..511][j=wg*COLS..+COLS) -> LDS ----
  {
    const int c  = tid & 15;          // 16B chunk within 256B row slice
    const int r0 = tid >> 4;          // 32 rows per pass
    for (int r = r0; r < L; r += NTHR / 16) {
      const float* gp = trans + (size_t)r * L + wg * COLS + c * 4;
      float* lp = ldsT + r * COLS + c * 4;
      async_copy_b128(gp, lp);
    }
  }
  wait_async_zero();
  __syncthreads();

  // ---- init: scores = feats[0]; publish slice, signal ----
  if (tid < COLS) {
    const int jG = wg * COLS + tid;
    scoresBuf[jG] = feats[jG];                 // buffer 0
  }
  fence_release_agent();
  __syncthreads();
  if (tid == 0)
    __hip_atomic_fetch_add(cnt, 1, __ATOMIC_RELEASE, __HIP_MEMORY_SCOPE_AGENT);

  const int jl = tid & (COLS - 1);   // local column
  const int tc = tid / COLS;         // chunk index (ascending i ranges)
  const int i0 = tc * CLEN;

#pragma unroll 1
  for (int t = 1; t < T_LEN; ++t) {
    // wait until all WGs finished step t-1: counter >= NWG * t
    if (tid == 0) {
      const int target = NWG * t;
      while (__hip_atomic_load(cnt, __ATOMIC_ACQUIRE, __HIP_MEMORY_SCOPE_AGENT) < target) {
        __builtin_amdgcn_s_sleep(1);
      }
    }
    __syncthreads();
    fence_acquire_agent();   // invalidate L0 path so fresh agent data visible per wave

    // stage previous scores (buffer (t+1)&1) into LDS, one f32 per thread
    async_copy_b32(scoresBuf + (size_t)((t + 1) & 1) * L + tid, ldsScores + tid);
    wait_async_zero();
    __syncthreads();

    // partial max-plus + argmax over i in [i0, i0+CLEN)
    float best = -__builtin_inff();
    int bidx = i0;
#pragma unroll
    for (int k4 = 0; k4 < CLEN / 4; ++k4) {
      const int i = i0 + k4 * 4;
      const float4 s4 = *(const float4*)(ldsScores + i);
      float m;
      m = s4.x + ldsT[(i + 0) * COLS + jl]; if (m > best) { best = m; bidx = i + 0; }
      m = s4.y + ldsT[(i + 1) * COLS + jl]; if (m > best) { best = m; bidx = i + 1; }
      m = s4.z + ldsT[(i + 2) * COLS + jl]; if (m > best) { best = m; bidx = i + 2; }
      m = s4.w + ldsT[(i + 3) * COLS + jl]; if (m > best) { best = m; bidx = i + 3; }
    }
    redV[tc * COLS + jl] = best;
    redI[tc * COLS + jl] = bidx;
    __syncthreads();

    // owners: combine CHUNKS partials (ascending i => strict '>' keeps first max)
    if (tid < COLS) {
      float bv = redV[tid];
      int   bi = redI[tid];
#pragma unroll
      for (int c2 = 1; c2 < CHUNKS; ++c2) {
        const float v = redV[c2 * COLS + tid];
        const int  ix = redI[c2 * COLS + tid];
        if (v > bv) { bv = v; bi = ix; }
      }
      const int jG = wg * COLS + tid;
      const float emis = feats[(size_t)t * L + jG];
      if (t + 16 < T_LEN)
        __builtin_prefetch(feats + (size_t)(t + 16) * L + jG, 0, 1);  // global_prefetch
      scoresBuf[(size_t)(t & 1) * L + jG] = emis + bv;
      bp[(size_t)(t - 1) * L + jG] = (unsigned short)bi;
    }
    fence_release_agent();
    __syncthreads();
    if (tid == 0)
      __hip_atomic_fetch_add(cnt, 1, __ATOMIC_RELEASE, __HIP_MEMORY_SCOPE_AGENT);
  }

  // ======================= backtrack =======================
  // wait for forward completion everywhere
  if (tid == 0) {
    const int target = NWG * T_LEN;   // NWG * (1 init + (T_LEN-1) steps)
    while (__hip_atomic_load(cnt, __ATOMIC_ACQUIRE, __HIP_MEMORY_SCOPE_AGENT) < target) {
      __builtin_amdgcn_s_sleep(1);
    }
  }
  __syncthreads();
  fence_acquire_agent();

  // ---- Phase A (all WGs): per-segment tag maps, 8 interleaved chains/thread.
  // segMap[s][e] = tag at time s*SEG when entering segment s (rows
  // [s*SEG, end_s)) with tag e at time end_s.  tag_k = bp[k][tag_{k+1}].
  {
    int tags[SEGS_PER_WG];
#pragma unroll
    for (int c = 0; c < SEGS_PER_WG; ++c) tags[c] = tid;
#pragma unroll 1
    for (int jj = 0; jj < SEG; ++jj) {
#pragma unroll
      for (int c = 0; c < SEGS_PER_WG; ++c) {
        const int s    = wg + c * NWG;
        const int endk = (s == NSEG - 1) ? (T_LEN - 1) : (s + 1) * SEG;
        const int k    = endk - 1 - jj;
        if (k >= s * SEG) tags[c] = (int)bp[(size_t)k * L + tags[c]];
      }
    }
#pragma unroll
    for (int c = 0; c < SEGS_PER_WG; ++c) {
      const int s = wg + c * NWG;
      segMap[(size_t)s * L + tid] = (unsigned short)tags[c];
    }
  }
  fence_release_agent();
  __syncthreads();
  if (tid == 0)
    __hip_atomic_fetch_add(cnt, 1, __ATOMIC_RELEASE, __HIP_MEMORY_SCOPE_AGENT);

  if (wg != 0) return;

  // ---- WG0: wait for all segment maps ----
  if (tid == 0) {
    const int target = NWG * T_LEN + NWG;
    while (__hip_atomic_load(cnt, __ATOMIC_ACQUIRE, __HIP_MEMORY_SCOPE_AGENT) < target) {
      __builtin_amdgcn_s_sleep(1);
    }
  }
  __syncthreads();
  fence_acquire_agent();

  // final scores -> LDS
  ldsScores[tid] = scoresBuf[(size_t)((T_LEN - 1) & 1) * L + tid];
  __syncthreads();

  // ---- Phase B: argmax + compose 64 segment maps (boundary tags in redI) ----
  if (tid == 0) {
    float best = ldsScores[0];
    int tag = 0;
    for (int i = 1; i < L; ++i) {
      const float v = ldsScores[i];
      if (v > best) { best = v; tag = i; }   // first-max like jnp.argmax
    }
    out[0] = best;
    out[1 + (T_LEN - 1)] = (float)tag;       // path at time T-1
    redI[NSEG] = tag;                        // boundary[s] = tag at time s*SEG
    for (int s = NSEG - 1; s >= 0; --s) {
      tag = (int)segMap[(size_t)s * L + tag];
      redI[s] = tag;
    }
  }
  __syncthreads();

  // ---- Phase C: 64 parallel segment walks emit the path ----
  if (tid < NSEG) {
    const int s    = tid;
    const int endk = (s == NSEG - 1) ? (T_LEN - 1) : (s + 1) * SEG;
    int tag = redI[s + 1];                   // tag at time endk
#pragma unroll 1
    for (int k = endk - 1; k >= s * SEG; --k) {
      tag = (int)bp[(size_t)k * L + tag];
      out[1 + k] = (float)tag;               // path at time k
    }
  }
}

extern "C" void kernel_launch(void* const* d_in, const int* in_sizes, int n_in,
                              void* d_out, int out_size, void* d_ws, size_t ws_size,
                              hipStream_t stream) {
  (void)in_sizes; (void)n_in; (void)out_size; (void)ws_size;
  const float* feats = (const float*)d_in[0];
  const float* trans = (const float*)d_in[1];
  float* out = (float*)d_out;

  uint8_t* ws = (uint8_t*)d_ws;
  int* cnt = (int*)ws;
  float* scoresBuf = (float*)(ws + 256);
  unsigned short* bp = (unsigned short*)(ws + 8192);
  unsigned short* segMap = (unsigned short*)(ws + 8192 + (size_t)(T_LEN - 1) * L * 2);
  // requires ws_size >= 8192 + 8191*512*2 + 64*512*2  (~8.46 MB)

  const size_t SMEM = (size_t)L * COLS * 4   // ldsT 128KB
                    + (size_t)L * 4          // ldsScores
                    + (size_t)CHUNKS * COLS * 4
                    + (size_t)(CHUNKS * COLS + 64) * 4;   // redI (+boundary slack)

  (void)hipFuncSetAttribute((const void*)crf_forward_kernel,
                            hipFuncAttributeMaxDynamicSharedMemorySize, (int)SMEM);

  crf_init_kernel<<<1, 1, 0, stream>>>(cnt);
  crf_forward_kernel<<<NWG, NTHR, SMEM, stream>>>(feats, trans, out, cnt,
                                                  scoresBuf, bp, segMap);
}